// ControlPointDeformation_56831007261103
// MI455X (gfx1250) — compile-verified
//
#include <hip/hip_runtime.h>
#include <hip/hip_bf16.h>

// Problem constants (fixed by setup_inputs)
#define BB   8
#define NN_  16384
#define MM   16384
#define FF   1024
#define HH   512
#define KC   128   // K_CTRL control points

typedef float v2f __attribute__((ext_vector_type(2)));
typedef float v8f __attribute__((ext_vector_type(8)));

// ---------------------------------------------------------------------------
// 1) Farthest point sampling: one block per batch, points held in registers.
//    Matches reference: cp_idx[0]=0, then 127 rounds of dmin-update + argmax.
// ---------------------------------------------------------------------------
__global__ __launch_bounds__(512) void fps_kernel(const float* __restrict__ src,
                                                  int* __restrict__ cp_idx) {
    const int b   = blockIdx.x;
    const int tid = threadIdx.x;
    const float* pb = src + (size_t)b * NN_ * 3;
    const int PPT = NN_ / 512;                 // 32 points per thread (strided)

    float px[PPT], py[PPT], pz[PPT], dmin[PPT];
#pragma unroll
    for (int i = 0; i < PPT; ++i) {
        int n = tid + i * 512;
        px[i] = pb[n * 3 + 0];
        py[i] = pb[n * 3 + 1];
        pz[i] = pb[n * 3 + 2];
        dmin[i] = INFINITY;
    }

    __shared__ float sval[512];
    __shared__ int   sidx[512];

    int far = 0;
    for (int k = 0; k < KC; ++k) {
        if (tid == 0) cp_idx[b * KC + k] = far;
        // broadcast centroid (all lanes read same address)
        float cx = pb[far * 3 + 0];
        float cy = pb[far * 3 + 1];
        float cz = pb[far * 3 + 2];

        float bv = -INFINITY; int bi = 0;
#pragma unroll
        for (int i = 0; i < PPT; ++i) {
            float dx = px[i] - cx, dy = py[i] - cy, dz = pz[i] - cz;
            float d = dx * dx + dy * dy + dz * dz;
            if (d < dmin[i]) dmin[i] = d;
            if (dmin[i] > bv) { bv = dmin[i]; bi = tid + i * 512; }  // ascending idx -> first max kept
        }
        sval[tid] = bv; sidx[tid] = bi;
        __syncthreads();
        for (int s = 256; s > 0; s >>= 1) {
            if (tid < s) {
                float v2 = sval[tid + s]; int i2 = sidx[tid + s];
                if (v2 > sval[tid] || (v2 == sval[tid] && i2 < sidx[tid])) {
                    sval[tid] = v2; sidx[tid] = i2;
                }
            }
            __syncthreads();
        }
        far = sidx[0];
        __syncthreads();   // protect sval/sidx before next round
    }
}

// ---------------------------------------------------------------------------
// 2) Per control point: gather coords/feats, NN search over targets (same
//    expression as reference: cp2 + tg2 - 2*cross), gather target feats.
//    One block per (b,k).
// ---------------------------------------------------------------------------
__global__ __launch_bounds__(256) void control_kernel(
        const float* __restrict__ src, const float* __restrict__ feats,
        const float* __restrict__ tgtp, const float* __restrict__ tgtf,
        const int* __restrict__ cp_idx,
        float* __restrict__ cp_ws, float* __restrict__ out_cp,
        float* __restrict__ mlp_in) {
    const int bk  = blockIdx.x;            // 0 .. B*K-1
    const int b   = bk / KC;
    const int tid = threadIdx.x;

    __shared__ float sminv[256];
    __shared__ int   smini[256];

    const int pidx = cp_idx[bk];
    const float* p = src + ((size_t)b * NN_ + pidx) * 3;
    const float cx = p[0], cy = p[1], cz = p[2];
    if (tid < 3) {
        float v = p[tid];
        cp_ws[bk * 3 + tid]  = v;
        out_cp[bk * 3 + tid] = v;
    }
    const float cp2 = cx * cx + cy * cy + cz * cz;

    // nearest neighbor over target points (first-min tie break, like argmin)
    const float* tb = tgtp + (size_t)b * MM * 3;
    float best = INFINITY; int bi = 0;
    for (int m = tid; m < MM; m += 256) {
        float tx = tb[m * 3 + 0], ty = tb[m * 3 + 1], tz = tb[m * 3 + 2];
        float tg2   = tx * tx + ty * ty + tz * tz;
        float cross = cx * tx + cy * ty + cz * tz;
        float s = cp2 + tg2 - 2.0f * cross;
        if (s < best) { best = s; bi = m; }
    }
    sminv[tid] = best; smini[tid] = bi;
    __syncthreads();
    for (int s = 128; s > 0; s >>= 1) {
        if (tid < s) {
            float v2 = sminv[tid + s]; int i2 = smini[tid + s];
            if (v2 < sminv[tid] || (v2 == sminv[tid] && i2 < smini[tid])) {
                sminv[tid] = v2; smini[tid] = i2;
            }
        }
        __syncthreads();
    }
    const int nn = smini[0];

    // gather features: mlp_in row = [control_feats | tgt_local_feats]
    const float* cf = feats + ((size_t)b * NN_ + pidx) * FF;
    const float* tf = tgtf  + ((size_t)b * MM  + nn)   * FF;
    float* row = mlp_in + (size_t)bk * (2 * FF);
    for (int j = tid; j < FF; j += 256) {
        row[j]      = cf[j];
        row[FF + j] = tf[j];
    }
}

// ---------------------------------------------------------------------------
// 3) fp32 WMMA GEMM: D = A(MxKd) @ W(KdxNd) + bias, one wave per 16x16 tile.
//    Uses V_WMMA_F32_16X16X4_F32 lane layouts (lanes 0-15: K=0..1, 16-31: K=2..3).
// ---------------------------------------------------------------------------
__global__ __launch_bounds__(32) void gemm_wmma_f32(
        const float* __restrict__ A, const float* __restrict__ W,
        const float* __restrict__ bias, float* __restrict__ D,
        int Kd, int Nd) {
    const int m0 = blockIdx.x * 16;
    const int n0 = blockIdx.y * 16;
    const int l  = threadIdx.x;
    const int lm   = l & 15;
    const int koff = (l < 16) ? 0 : 2;

    const float bv = bias[n0 + lm];
    v8f acc;
#pragma unroll
    for (int r = 0; r < 8; ++r) acc[r] = bv;   // C[m][n] = bias[n]

    const float* arow = A + (size_t)(m0 + lm) * Kd;
    for (int kk = 0; kk < Kd; kk += 4) {
        v2f a, bb;
        a[0]  = arow[kk + koff];
        a[1]  = arow[kk + koff + 1];
        bb[0] = W[(size_t)(kk + koff)     * Nd + n0 + lm];
        bb[1] = W[(size_t)(kk + koff + 1) * Nd + n0 + lm];
        acc = __builtin_amdgcn_wmma_f32_16x16x4_f32(
                  false, a, false, bb, (short)0, acc, false, false);
    }

    const int rbase = m0 + ((l < 16) ? 0 : 8);
#pragma unroll
    for (int r = 0; r < 8; ++r)
        D[(size_t)(rbase + r) * Nd + n0 + lm] = acc[r];
}

// ---------------------------------------------------------------------------
// 4) BatchNorm (training-mode, biased var) statistics: one block per column.
// ---------------------------------------------------------------------------
__global__ __launch_bounds__(256) void bnstats_kernel(
        const float* __restrict__ h, float* __restrict__ mu,
        float* __restrict__ rstd, int Mrows, int Ncols) {
    const int c = blockIdx.x, tid = threadIdx.x;
    __shared__ float sred[256];
    const int cnt = Mrows / 256;   // 4
    float vals[4];
    float s = 0.0f;
    for (int i = 0; i < cnt; ++i) {
        vals[i] = h[(size_t)(tid + i * 256) * Ncols + c];
        s += vals[i];
    }
    sred[tid] = s; __syncthreads();
    for (int st = 128; st > 0; st >>= 1) {
        if (tid < st) sred[tid] += sred[tid + st];
        __syncthreads();
    }
    const float m = sred[0] / (float)Mrows;
    __syncthreads();
    float sq = 0.0f;
    for (int i = 0; i < cnt; ++i) { float d = vals[i] - m; sq += d * d; }
    sred[tid] = sq; __syncthreads();
    for (int st = 128; st > 0; st >>= 1) {
        if (tid < st) sred[tid] += sred[tid + st];
        __syncthreads();
    }
    if (tid == 0) {
        mu[c]   = m;
        rstd[c] = rsqrtf(sred[0] / (float)Mrows + 1e-5f);
    }
}

__global__ __launch_bounds__(256) void bnapply_kernel(
        const float* __restrict__ h, const float* __restrict__ mu,
        const float* __restrict__ rstd, const float* __restrict__ g,
        const float* __restrict__ beta, float* __restrict__ o,
        int total, int Ncols) {
    int i = blockIdx.x * 256 + threadIdx.x;
    if (i >= total) return;
    int c = i % Ncols;
    float v = g[c] * (h[i] - mu[c]) * rstd[c] + beta[c];
    o[i] = v > 0.0f ? v : 0.0f;
}

// ---------------------------------------------------------------------------
// 5) Final tiny layer: w = h @ W3(512x3) + b3. One thread per row.
// ---------------------------------------------------------------------------
__global__ __launch_bounds__(128) void gemm3_kernel(
        const float* __restrict__ h, const float* __restrict__ W3,
        const float* __restrict__ b3, float* __restrict__ w_ws,
        float* __restrict__ out_w, int Mrows) {
    int r = blockIdx.x * 128 + threadIdx.x;
    if (r >= Mrows) return;
    const float* hr = h + (size_t)r * HH;
    float a0 = b3[0], a1 = b3[1], a2 = b3[2];
    for (int j = 0; j < HH; ++j) {
        float v = hr[j];
        a0 += v * W3[j * 3 + 0];
        a1 += v * W3[j * 3 + 1];
        a2 += v * W3[j * 3 + 2];
    }
    w_ws[r * 3 + 0] = a0; out_w[r * 3 + 0] = a0;
    w_ws[r * 3 + 1] = a1; out_w[r * 3 + 1] = a1;
    w_ws[r * 3 + 2] = a2; out_w[r * 3 + 2] = a2;
}

// ---------------------------------------------------------------------------
// 6) RBF-softmax interpolation: grid (N/256, B); cp & w tiles in LDS.
//    softmax(-sqd / (2*0.5^2)) == softmax(-2*sqd), with max-subtraction.
// ---------------------------------------------------------------------------
__global__ __launch_bounds__(256) void rbf_kernel(
        const float* __restrict__ src, const float* __restrict__ cp,
        const float* __restrict__ w, float* __restrict__ out_o,
        float* __restrict__ out_d) {
    const int b   = blockIdx.y;
    const int tid = threadIdx.x;
    const int n   = blockIdx.x * 256 + tid;

    __shared__ float scp[KC * 3];
    __shared__ float sw[KC * 3];
    for (int i = tid; i < KC * 3; i += 256) {
        scp[i] = cp[(size_t)b * KC * 3 + i];
        sw[i]  = w [(size_t)b * KC * 3 + i];
    }
    __syncthreads();

    const float* p = src + ((size_t)b * NN_ + n) * 3;
    const float x = p[0], y = p[1], z = p[2];

    float smax = -INFINITY;
    for (int k = 0; k < KC; ++k) {
        float dx = x - scp[k * 3 + 0];
        float dy = y - scp[k * 3 + 1];
        float dz = z - scp[k * 3 + 2];
        float s = -2.0f * (dx * dx + dy * dy + dz * dz);
        if (s > smax) smax = s;
    }
    float se = 0.0f, ax = 0.0f, ay = 0.0f, az = 0.0f;
    for (int k = 0; k < KC; ++k) {
        float dx = x - scp[k * 3 + 0];
        float dy = y - scp[k * 3 + 1];
        float dz = z - scp[k * 3 + 2];
        float s = -2.0f * (dx * dx + dy * dy + dz * dz);
        float e = expf(s - smax);
        se += e;
        ax += e * sw[k * 3 + 0];
        ay += e * sw[k * 3 + 1];
        az += e * sw[k * 3 + 2];
    }
    const float inv = 1.0f / se;
    const float dx = ax * inv, dy = ay * inv, dz = az * inv;
    const size_t o = ((size_t)b * NN_ + n) * 3;
    out_d[o + 0] = dx;     out_d[o + 1] = dy;     out_d[o + 2] = dz;
    out_o[o + 0] = x + dx; out_o[o + 1] = y + dy; out_o[o + 2] = z + dz;
}

// ---------------------------------------------------------------------------
extern "C" void kernel_launch(void* const* d_in, const int* in_sizes, int n_in,
                              void* d_out, int out_size, void* d_ws, size_t ws_size,
                              hipStream_t stream) {
    const float* src   = (const float*)d_in[0];   // (B,N,3)
    const float* feats = (const float*)d_in[1];   // (B,N,F)
    const float* tgtp  = (const float*)d_in[2];   // (B,M,3)
    const float* tgtf  = (const float*)d_in[3];   // (B,M,F)
    const float* W1    = (const float*)d_in[4];   // (2F,H)
    const float* b1    = (const float*)d_in[5];
    const float* g1    = (const float*)d_in[6];
    const float* beta1 = (const float*)d_in[7];
    const float* W2    = (const float*)d_in[8];   // (H,H)
    const float* b2    = (const float*)d_in[9];
    const float* g2    = (const float*)d_in[10];
    const float* beta2 = (const float*)d_in[11];
    const float* W3    = (const float*)d_in[12];  // (H,3)
    const float* b3    = (const float*)d_in[13];

    float* out = (float*)d_out;
    float* out_output = out;                                   // (B,N,3)
    float* out_delta  = out + (size_t)BB * NN_ * 3;            // (B,N,3)
    float* out_cp     = out + (size_t)2 * BB * NN_ * 3;        // (B,K,3)
    float* out_w      = out_cp + (size_t)BB * KC * 3;          // (B,K,3)

    // workspace layout (bytes)
    char* ws = (char*)d_ws;
    int*   cp_idx = (int*)(ws + 0);                  //  4 KB
    float* cp_ws  = (float*)(ws + 8192);             // 12 KB
    float* w_ws   = (float*)(ws + 20480);            // 12 KB
    float* mu1    = (float*)(ws + 32768);
    float* rstd1  = (float*)(ws + 34816);
    float* mu2    = (float*)(ws + 36864);
    float* rstd2  = (float*)(ws + 38912);
    float* mlp_in = (float*)(ws + 40960);            // 1024*2048*4 = 8 MB (reused for h2n)
    float* hA     = (float*)(ws + 40960 + (size_t)1024 * 2048 * 4);          // 2 MB (h1 / h2)
    float* hB     = (float*)(ws + 40960 + (size_t)1024 * 2048 * 4 + (size_t)1024 * 512 * 4); // 2 MB (h1n)
    float* h2n    = mlp_in;  // reuse: mlp_in no longer needed after GEMM2

    const int MROWS = BB * KC;   // 1024

    // 1. FPS
    fps_kernel<<<BB, 512, 0, stream>>>(src, cp_idx);

    // 2. control point gather + NN + feature assembly
    control_kernel<<<MROWS, 256, 0, stream>>>(src, feats, tgtp, tgtf, cp_idx,
                                              cp_ws, out_cp, mlp_in);

    // 3. layer 1: (1024 x 2048) @ (2048 x 512) + b1  (fp32 WMMA)
    gemm_wmma_f32<<<dim3(MROWS / 16, HH / 16), 32, 0, stream>>>(
        mlp_in, W1, b1, hA, 2 * FF, HH);
    bnstats_kernel<<<HH, 256, 0, stream>>>(hA, mu1, rstd1, MROWS, HH);
    bnapply_kernel<<<(MROWS * HH + 255) / 256, 256, 0, stream>>>(
        hA, mu1, rstd1, g1, beta1, hB, MROWS * HH, HH);

    // 4. layer 2: (1024 x 512) @ (512 x 512) + b2  (fp32 WMMA)
    gemm_wmma_f32<<<dim3(MROWS / 16, HH / 16), 32, 0, stream>>>(
        hB, W2, b2, hA, HH, HH);
    bnstats_kernel<<<HH, 256, 0, stream>>>(hA, mu2, rstd2, MROWS, HH);
    bnapply_kernel<<<(MROWS * HH + 255) / 256, 256, 0, stream>>>(
        hA, mu2, rstd2, g2, beta2, h2n, MROWS * HH, HH);

    // 5. layer 3 -> control displacements w
    gemm3_kernel<<<(MROWS + 127) / 128, 128, 0, stream>>>(h2n, W3, b3, w_ws,
                                                          out_w, MROWS);

    // 6. RBF-softmax interpolation to all points
    rbf_kernel<<<dim3(NN_ / 256, BB), 256, 0, stream>>>(src, cp_ws, w_ws,
                                                        out_output, out_delta);
}